// FastAttention_69380901700183
// MI455X (gfx1250) — compile-verified
//
#include <hip/hip_runtime.h>
#include <math.h>

typedef __attribute__((ext_vector_type(2))) float v2f;
typedef __attribute__((ext_vector_type(8))) float v8f;

#define Bb 64
#define Ss 2048
#define Dd 512
#define CHUNKS 16

// ---------------------------------------------------------------------------
// Pass over x: partial column-sum (optionally weighted by per-(b,s) weight w).
// grid (B, CHUNKS), block Dd. Coalesced 2KB row loads.
// ---------------------------------------------------------------------------
__global__ void fa_colsum_partial(const float* __restrict__ x,
                                  const float* __restrict__ w,  // nullable
                                  float* __restrict__ part) {
    const int b = blockIdx.x, chunk = blockIdx.y;
    const int d = threadIdx.x;
    const int per = Ss / CHUNKS;
    const float* xb = x + ((size_t)b * Ss + (size_t)chunk * per) * Dd + d;
    float acc = 0.f;
    if (w) {
        const float* wb = w + (size_t)b * Ss + (size_t)chunk * per;
#pragma unroll 8
        for (int s = 0; s < per; ++s) acc += wb[s] * xb[(size_t)s * Dd];
    } else {
#pragma unroll 8
        for (int s = 0; s < per; ++s) acc += xb[(size_t)s * Dd];
    }
    part[((size_t)b * CHUNKS + chunk) * Dd + d] = acc;
}

// grid B, block Dd: reduce CHUNKS partials, scale.
__global__ void fa_reduce_chunks(const float* __restrict__ part,
                                 float* __restrict__ out, float scale) {
    const int b = blockIdx.x, d = threadIdx.x;
    float acc = 0.f;
#pragma unroll
    for (int c = 0; c < CHUNKS; ++c)
        acc += part[((size_t)b * CHUNKS + c) * Dd + d];
    out[(size_t)b * Dd + d] = acc * scale;
}

// ---------------------------------------------------------------------------
// WMMA fp32 GEMM: Out[M,N] = A[M,K] x Bmat[K,N] (+ bias[N]).
// BT=true : Bmat[k][n] = W[n*K + k]  (einsum '...d,ed->...e' orientation)
// BT=false: Bmat[k][n] = W[k*N + n]  (einsum 'be,ed->bd' orientation)
// One wave per 16x16 tile; V_WMMA_F32_16X16X4_F32 over K in steps of 4.
// A layout: lane<16 -> row m0+lane, K = k0,k0+1 ; lane>=16 -> K = k0+2,k0+3.
// ---------------------------------------------------------------------------
template <bool BT>
__global__ void fa_gemm_wmma(const float* __restrict__ A,
                             const float* __restrict__ W,
                             const float* __restrict__ bias,  // nullable
                             float* __restrict__ Out,
                             int M, int K, int N) {
    const int wave = (blockIdx.x * blockDim.x + threadIdx.x) >> 5;
    const int lane = threadIdx.x & 31;
    const int ntn  = N >> 4;
    const int m0   = (wave / ntn) << 4;
    const int n0   = (wave % ntn) << 4;
    const int half = lane >> 4;       // 0 or 1
    const int l15  = lane & 15;
    const int koff = half << 1;       // 0 or 2

    v8f acc = {};
    const float* arow = A + (size_t)(m0 + l15) * K;
#pragma unroll 4
    for (int k0 = 0; k0 < K; k0 += 4) {
        v2f a = *(const v2f*)(arow + k0 + koff);
        v2f b;
        if (BT) {
            b = *(const v2f*)(W + (size_t)(n0 + l15) * K + k0 + koff);
        } else {
            b[0] = W[(size_t)(k0 + koff) * N + n0 + l15];
            b[1] = W[(size_t)(k0 + koff + 1) * N + n0 + l15];
        }
        acc = __builtin_amdgcn_wmma_f32_16x16x4_f32(
            /*neg_a=*/false, a, /*neg_b=*/false, b,
            /*c_mod=*/(short)0, acc, /*reuse_a=*/false, /*reuse_b=*/false);
    }
    const float bv = bias ? bias[n0 + l15] : 0.f;
#pragma unroll
    for (int r = 0; r < 8; ++r) {
        const int row = m0 + r + half * 8;          // VGPR r: M=r (lo) / r+8 (hi)
        Out[(size_t)row * N + n0 + l15] = acc[r] + bv;
    }
}

// grid B, block 256: c[b] = dot(q[b,:], bk)
__global__ void fa_dot_bias(const float* __restrict__ q,
                            const float* __restrict__ bk,
                            float* __restrict__ c) {
    __shared__ float red[256];
    const int b = blockIdx.x, t = threadIdx.x;
    float acc = 0.f;
    for (int i = t; i < Dd; i += 256) acc += q[(size_t)b * Dd + i] * bk[i];
    red[t] = acc;
    __syncthreads();
    for (int s = 128; s > 0; s >>= 1) {
        if (t < s) red[t] += red[t + s];
        __syncthreads();
    }
    if (t == 0) c[b] = red[0];
}

// One wave per (b,s): scores[b,s] = dot(x[b,s,:], u[b,:]) + c[b]
__global__ void fa_scores(const float* __restrict__ x,
                          const float* __restrict__ u,
                          const float* __restrict__ c,
                          float* __restrict__ scores) {
    const int gwave = (blockIdx.x * blockDim.x + threadIdx.x) >> 5;
    const int lane  = threadIdx.x & 31;
    const int b = gwave / Ss, s = gwave % Ss;
    const float4* xr = (const float4*)(x + ((size_t)b * Ss + s) * Dd);
    const float4* ur = (const float4*)(u + (size_t)b * Dd);
    float acc = 0.f;
#pragma unroll
    for (int i = lane; i < Dd / 4; i += 32) {
        float4 xv = xr[i], uv = ur[i];
        acc += xv.x * uv.x + xv.y * uv.y + xv.z * uv.z + xv.w * uv.w;
    }
#pragma unroll
    for (int off = 16; off > 0; off >>= 1) acc += __shfl_xor(acc, off, 32);
    if (lane == 0) scores[(size_t)b * Ss + s] = acc + c[b];
}

// grid B, block 256: in-place softmax over S
__global__ void fa_softmax(float* __restrict__ scores) {
    __shared__ float red[256];
    const int b = blockIdx.x, t = threadIdx.x;
    float* row = scores + (size_t)b * Ss;
    float mx = -INFINITY;
    for (int i = t; i < Ss; i += 256) mx = fmaxf(mx, row[i]);
    red[t] = mx;
    __syncthreads();
    for (int s = 128; s > 0; s >>= 1) {
        if (t < s) red[t] = fmaxf(red[t], red[t + s]);
        __syncthreads();
    }
    mx = red[0];
    __syncthreads();
    float sum = 0.f;
    for (int i = t; i < Ss; i += 256) {
        float e = __expf(row[i] - mx);
        row[i] = e;
        sum += e;
    }
    red[t] = sum;
    __syncthreads();
    for (int s = 128; s > 0; s >>= 1) {
        if (t < s) red[t] += red[t + s];
        __syncthreads();
    }
    const float inv = 1.0f / red[0];
    for (int i = t; i < Ss; i += 256) row[i] *= inv;
}

// ---------------------------------------------------------------------------
extern "C" void kernel_launch(void* const* d_in, const int* in_sizes, int n_in,
                              void* d_out, int out_size, void* d_ws, size_t ws_size,
                              hipStream_t stream) {
    const float* x  = (const float*)d_in[0];
    const float* Wq = (const float*)d_in[1];
    const float* bq = (const float*)d_in[2];
    const float* Wk = (const float*)d_in[3];
    const float* bk = (const float*)d_in[4];
    const float* Wv = (const float*)d_in[5];
    const float* bv = (const float*)d_in[6];
    float* out = (float*)d_out;

    float* ws   = (float*)d_ws;
    float* part = ws;                               // B*CHUNKS*D
    float* m    = part + (size_t)Bb * CHUNKS * Dd;  // B*D  (mean pooled x)
    float* q    = m + (size_t)Bb * Dd;              // B*D
    float* u    = q + (size_t)Bb * Dd;              // B*D  (q @ Wk)
    float* p    = u + (size_t)Bb * Dd;              // B*D  (alpha-pooled x)
    float* c    = p + (size_t)Bb * Dd;              // B
    float* sc   = c + Bb;                           // B*S

    const int gemm_tiles  = (Bb / 16) * (Dd / 16);  // 128 tiles
    const int gemm_blocks = gemm_tiles / 8;         // 8 waves/block

    // 1) m = mean_s(x)
    fa_colsum_partial<<<dim3(Bb, CHUNKS), Dd, 0, stream>>>(x, nullptr, part);
    fa_reduce_chunks<<<Bb, Dd, 0, stream>>>(part, m, 1.0f / (float)Ss);
    // 2) q = m @ Wq^T + bq
    fa_gemm_wmma<true><<<gemm_blocks, 256, 0, stream>>>(m, Wq, bq, q, Bb, Dd, Dd);
    // 3) u = q @ Wk ; c = q . bk
    fa_gemm_wmma<false><<<gemm_blocks, 256, 0, stream>>>(q, Wk, nullptr, u, Bb, Dd, Dd);
    fa_dot_bias<<<Bb, 256, 0, stream>>>(q, bk, c);
    // 4) scores[b,s] = x[b,s,:].u[b,:] + c[b]   (big pass over x)
    fa_scores<<<(Bb * Ss) / 8, 256, 0, stream>>>(x, u, c, sc);
    // 5) softmax over s (in place -> alpha)
    fa_softmax<<<Bb, 256, 0, stream>>>(sc);
    // 6) p = sum_s alpha[b,s] * x[b,s,:]        (big pass over x)
    fa_colsum_partial<<<dim3(Bb, CHUNKS), Dd, 0, stream>>>(x, sc, part);
    fa_reduce_chunks<<<Bb, Dd, 0, stream>>>(part, p, 1.0f);
    // 7) out = p @ Wv^T + bv
    fa_gemm_wmma<true><<<gemm_blocks, 256, 0, stream>>>(p, Wv, bv, out, Bb, Dd, Dd);
}